// GeneralizedMHDPAttentionModule_84722524881253
// MI455X (gfx1250) — compile-verified
//
#include <hip/hip_runtime.h>
#include <hip/hip_bf16.h>

typedef __attribute__((ext_vector_type(2))) float v2f;
typedef __attribute__((ext_vector_type(8))) float v8f;

namespace {

constexpr int kHeads = 16;
constexpr int kDqk   = 64;
constexpr int kDv    = 64;
constexpr int kDin   = 1024;
constexpr int kDout  = 1024;
constexpr int kB     = 2;
constexpr int kN     = 2048;
constexpr float kEps = 1e-8f;
constexpr float kSF  = 0.125f; // 1/sqrt(64)

constexpr int TILE = 64;  // output tile edge
constexpr int KCH  = 16;  // K chunk per LDS stage
constexpr int KPAD = 20;  // LDS row pitch in floats (bank-conflict-free, 16B aligned float4 slots)

// Generic fp32 WMMA GEMM: C[M x Nc] = alpha * A[M x Kd] * B[Kd x Nc]
// BT = true  -> B is stored as Bsrc[n*ldb + k] (i.e. we multiply by Bsrc^T)
// BT = false -> B is row-major Bsrc[k*ldb + n]
// Batched via blockIdx.z: z = bb*nH + hh, element offsets bb*?SB + hh*?SH.
// Requires M,Nc multiples of 64 and Kd multiple of 16 (true for all uses here).
template <bool BT>
__global__ __launch_bounds__(256) void gemm_wmma_f32(
    const float* __restrict__ A, const float* __restrict__ B, float* __restrict__ C,
    int Kd, int lda, int ldb, int ldc, float alpha, int nH,
    long long aSB, long long aSH, long long bSB, long long bSH,
    long long cSB, long long cSH)
{
  __shared__ float As[TILE][KPAD];
  __shared__ float Bt[TILE][KPAD];  // stored as [n][k]

  const int z  = blockIdx.z;
  const int bb = z / nH;
  const int hh = z - bb * nH;
  const float* Ab = A + bb * aSB + hh * aSH;
  const float* Bb = B + bb * bSB + hh * bSH;
  float*       Cb = C + bb * cSB + hh * cSH;

  const int tileN = blockIdx.x * TILE;
  const int tileM = blockIdx.y * TILE;
  const int tid   = threadIdx.x;
  const int lane  = tid & 31;
  const int wave  = tid >> 5;

  // 16 subtiles (4x4 grid of 16x16); wave w handles (si, sj0) and (si, sj1).
  const int si   = wave >> 1;
  const int sj0  = (wave & 1) * 2;
  const int sj1  = sj0 + 1;
  const int fm   = lane & 15;          // row (A) / col (B) within fragment
  const int kh   = (lane >> 4) * 2;    // K sub-offset per lane half (K=0/1 vs K=2/3)
  const int rofs = (lane >> 4) * 8;    // C/D: lanes 16-31 hold rows M+8..M+15

  v8f acc0 = {};
  v8f acc1 = {};

  const int ldRow = tid >> 2;          // 0..63
  const int ldK   = (tid & 3) * 4;     // 0,4,8,12
  const int bKk   = tid >> 4;          // 0..15  (BT=false path)
  const int bJ4   = (tid & 15) * 4;    // 0..60

  for (int k0 = 0; k0 < Kd; k0 += KCH) {
    // Stage A tile (64 rows x 16 k), k contiguous -> float4 per thread.
    const float4 av = *(const float4*)(Ab + (long long)(tileM + ldRow) * lda + (k0 + ldK));
    *(float4*)&As[ldRow][ldK] = av;
    if (BT) {
      // B^T: Bsrc[n][k] row-major -> same pattern as A.
      const float4 bv = *(const float4*)(Bb + (long long)(tileN + ldRow) * ldb + (k0 + ldK));
      *(float4*)&Bt[ldRow][ldK] = bv;
    } else {
      // B row-major [k][n]: load float4 along n, scatter-transpose into LDS.
      const float4 bv = *(const float4*)(Bb + (long long)(k0 + bKk) * ldb + (tileN + bJ4));
      Bt[bJ4 + 0][bKk] = bv.x;
      Bt[bJ4 + 1][bKk] = bv.y;
      Bt[bJ4 + 2][bKk] = bv.z;
      Bt[bJ4 + 3][bKk] = bv.w;
    }
    __syncthreads();

#pragma unroll
    for (int kk = 0; kk < KCH; kk += 4) {
      // A 16x4 fragment (ISA 7.12.2): VGPR0 = K=0 (lanes 0-15) / K=2 (lanes 16-31), VGPR1 = K=1/K=3.
      v2f a, b0, b1;
      a[0]  = As[si * 16 + fm][kk + kh];
      a[1]  = As[si * 16 + fm][kk + kh + 1];
      b0[0] = Bt[sj0 * 16 + fm][kk + kh];
      b0[1] = Bt[sj0 * 16 + fm][kk + kh + 1];
      b1[0] = Bt[sj1 * 16 + fm][kk + kh];
      b1[1] = Bt[sj1 * 16 + fm][kk + kh + 1];
      acc0 = __builtin_amdgcn_wmma_f32_16x16x4_f32(false, a, false, b0, (short)0, acc0, false, false);
      acc1 = __builtin_amdgcn_wmma_f32_16x16x4_f32(false, a, false, b1, (short)0, acc1, false, false);
    }
    __syncthreads();
  }

  // C/D 16x16 layout: VGPR r -> M = r (lanes 0-15) or r+8 (lanes 16-31), N = lane&15.
#pragma unroll
  for (int r = 0; r < 8; ++r) {
    const int row = tileM + si * 16 + rofs + r;
    Cb[(long long)row * ldc + (tileN + sj0 * 16 + fm)] = alpha * acc0[r];
    Cb[(long long)row * ldc + (tileN + sj1 * 16 + fm)] = alpha * acc1[r];
  }
}

// softmax over the *query* axis n (axis=-2): per (b,h,m) column of attn[b,h,n,m].
// 256 consecutive m per block -> fully coalesced column sweeps.
__global__ __launch_bounds__(256) void col_softmax_kernel(float* __restrict__ attn)
{
  const int bh = blockIdx.y;
  const int m  = blockIdx.x * blockDim.x + threadIdx.x;
  float* base = attn + (size_t)bh * kN * kN + m;

  float mx = -3.4e38f;
  float s  = 0.f;
  for (int n = 0; n < kN; ++n) {
    const float v  = base[(size_t)n * kN];
    const float nm = fmaxf(mx, v);
    s  = s * __expf(mx - nm) + __expf(v - nm);
    mx = nm;
  }
  const float inv = 1.0f / s;
  for (int n = 0; n < kN; ++n) {
    const size_t idx = (size_t)n * kN;
    base[idx] = __expf(base[idx] - mx) * inv;
  }
}

// renormalize over keys m: attn / (sum_m attn + eps). One block per row.
__global__ __launch_bounds__(256) void row_renorm_kernel(float* __restrict__ attn)
{
  __shared__ float red[256];
  float* base = attn + (size_t)blockIdx.x * kN;

  float s = 0.f;
  for (int m = threadIdx.x; m < kN; m += 256) s += base[m];
  red[threadIdx.x] = s;
  __syncthreads();
#pragma unroll
  for (int off = 128; off > 0; off >>= 1) {
    if (threadIdx.x < off) red[threadIdx.x] += red[threadIdx.x + off];
    __syncthreads();
  }
  const float inv = 1.0f / (red[0] + kEps);
  for (int m = threadIdx.x; m < kN; m += 256) base[m] *= inv;
}

}  // namespace

extern "C" void kernel_launch(void* const* d_in, const int* in_sizes, int n_in,
                              void* d_out, int out_size, void* d_ws, size_t ws_size,
                              hipStream_t stream)
{
  const float* X  = (const float*)d_in[0];  // (B,N,Din)
  const float* Wq = (const float*)d_in[1];  // (Din, H*dq)
  const float* Wk = (const float*)d_in[2];
  const float* Wv = (const float*)d_in[3];
  const float* Wo = (const float*)d_in[4];  // (H*dv, Dout)

  float* updates = (float*)d_out;                          // (B,N,Dout)
  float* attn    = updates + (size_t)kB * kN * kDout;      // (B,H,N,N)

  // Workspace: Q,K,V,O — 4 x 16 MB = 64 MB.
  const size_t projElems = (size_t)kB * kN * (kHeads * kDqk);
  float* Q  = (float*)d_ws;
  float* Kb = Q + projElems;
  float* Vb = Kb + projElems;
  float* O  = Vb + projElems;

  const dim3 blk(256);
  const int Mrows = kB * kN;  // 4096

  // 1) Q/K/V projections: (4096x1024) * (1024x1024)
  {
    dim3 grid((kHeads * kDqk) / TILE, Mrows / TILE, 1);
    gemm_wmma_f32<false><<<grid, blk, 0, stream>>>(X, Wq, Q, kDin, kDin, kHeads * kDqk,
                                                   kHeads * kDqk, 1.0f, 1, 0, 0, 0, 0, 0, 0);
    gemm_wmma_f32<false><<<grid, blk, 0, stream>>>(X, Wk, Kb, kDin, kDin, kHeads * kDqk,
                                                   kHeads * kDqk, 1.0f, 1, 0, 0, 0, 0, 0, 0);
    gemm_wmma_f32<false><<<grid, blk, 0, stream>>>(X, Wv, Vb, kDin, kDin, kHeads * kDv,
                                                   kHeads * kDv, 1.0f, 1, 0, 0, 0, 0, 0, 0);
  }

  // 2) scores: per (b,h)  S = SF * Q_h (2048x64) * K_h^T -> attn[b,h,:,:]
  {
    dim3 grid(kN / TILE, kN / TILE, kB * kHeads);
    const long long qSB = (long long)kN * (kHeads * kDqk);  // per-batch elems
    const long long aSB = (long long)kHeads * kN * kN;
    const long long aSH = (long long)kN * kN;
    gemm_wmma_f32<true><<<grid, blk, 0, stream>>>(Q, Kb, attn, kDqk,
                                                  kHeads * kDqk, kHeads * kDqk, kN, kSF, kHeads,
                                                  qSB, (long long)kDqk, qSB, (long long)kDqk,
                                                  aSB, aSH);
  }

  // 3) softmax over query axis n (per column m)
  {
    dim3 grid(kN / 256, kB * kHeads);
    col_softmax_kernel<<<grid, blk, 0, stream>>>(attn);
  }

  // 4) renormalize over key axis m (per row n)
  {
    dim3 grid(kB * kHeads * kN);
    row_renorm_kernel<<<grid, blk, 0, stream>>>(attn);
  }

  // 5) O_h = attn_h (2048x2048) * V_h (2048x64) -> O[b,n,h,v]
  {
    dim3 grid(kDv / TILE, kN / TILE, kB * kHeads);
    const long long aSB = (long long)kHeads * kN * kN;
    const long long aSH = (long long)kN * kN;
    const long long vSB = (long long)kN * (kHeads * kDv);
    gemm_wmma_f32<false><<<grid, blk, 0, stream>>>(attn, Vb, O, kN,
                                                   kN, kHeads * kDv, kHeads * kDv, 1.0f, kHeads,
                                                   aSB, aSH, vSB, (long long)kDv,
                                                   vSB, (long long)kDv);
  }

  // 6) updates = O (4096x1024) * Wo (1024x1024)
  {
    dim3 grid(kDout / TILE, Mrows / TILE, 1);
    gemm_wmma_f32<false><<<grid, blk, 0, stream>>>(O, Wo, updates, kHeads * kDv,
                                                   kHeads * kDv, kDout, kDout, 1.0f, 1,
                                                   0, 0, 0, 0, 0, 0);
  }
}